// ST_GCN_21406117004185
// MI455X (gfx1250) — compile-verified
//
#include <hip/hip_runtime.h>
#include <hip/hip_bf16.h>

typedef __attribute__((ext_vector_type(2))) float v2f;
typedef __attribute__((ext_vector_type(8))) float v8f;

static constexpr int  Bn = 8;
static constexpr int  Nn = 5000;
static constexpr int  Fc = 8;
static constexpr int  Hc = 64;
static constexpr int  Ec = 80000;
static constexpr int  T1 = 22;               // T - KS + 1
static constexpr int  T2 = 20;               // T1 - KS + 1
static constexpr long M1 = (long)Bn * T1 * Nn;   // 880000 (div by 128)
static constexpr long M2 = (long)Bn * T2 * Nn;   // 800000 (div by 128)
static constexpr long S1 = M1 * Hc;              // 56,320,000 floats

__device__ __forceinline__ v8f wmma4(v2f a, v2f b, v8f c) {
  // D(16x16,f32) += A(16x4,f32) * B(4x16,f32)  -> v_wmma_f32_16x16x4_f32
  return __builtin_amdgcn_wmma_f32_16x16x4_f32(
      /*neg_a=*/false, a, /*neg_b=*/false, b,
      /*c_mod=*/(short)0, c, /*reuse_a=*/false, /*reuse_b=*/false);
}

// ---------------------------------------------------------------- utilities
__global__ void zero_f_kernel(float* p, long n) {
  long i = (long)blockIdx.x * blockDim.x + threadIdx.x;
  if (i < n) p[i] = 0.f;
}
__global__ void zero_i_kernel(int* p, long n) {
  long i = (long)blockIdx.x * blockDim.x + threadIdx.x;
  if (i < n) p[i] = 0;
}

// ------------------------------------------------- cheb norm + CSR building
__global__ void deg_kernel(const int* __restrict__ src, const float* __restrict__ w,
                           float* __restrict__ deg) {
  int e = blockIdx.x * blockDim.x + threadIdx.x;
  if (e < Ec) atomicAdd(&deg[src[e]], w[e]);
}
__global__ void norm_kernel(const int* __restrict__ src, const int* __restrict__ dst,
                            const float* __restrict__ w, const float* __restrict__ deg,
                            float* __restrict__ nrm) {
  int e = blockIdx.x * blockDim.x + threadIdx.x;
  if (e >= Ec) return;
  float ds = deg[src[e]], dd = deg[dst[e]];
  float is = ds > 0.f ? rsqrtf(ds) : 0.f;
  float id = dd > 0.f ? rsqrtf(dd) : 0.f;
  nrm[e] = -is * w[e] * id;
}
__global__ void count_kernel(const int* __restrict__ dst, int* __restrict__ cnt) {
  int e = blockIdx.x * blockDim.x + threadIdx.x;
  if (e < Ec) atomicAdd(&cnt[dst[e]], 1);
}
__global__ void scan_kernel(const int* __restrict__ cnt, int* __restrict__ rowptr) {
  if (threadIdx.x == 0 && blockIdx.x == 0) {
    int s = 0; rowptr[0] = 0;
    for (int n = 0; n < Nn; ++n) { s += cnt[n]; rowptr[n + 1] = s; }
  }
}
__global__ void fill_kernel(const int* __restrict__ dst, const int* __restrict__ rowptr,
                            int* __restrict__ cursor, int* __restrict__ eid) {
  int e = blockIdx.x * blockDim.x + threadIdx.x;
  if (e >= Ec) return;
  int d = dst[e];
  int pos = atomicAdd(&cursor[d], 1);
  eid[rowptr[d] + pos] = e;
}

// ------------------------------------------- sparse propagation (CSR gather)
// out[bt,n,c] = alpha * sum_{e: dst=n} nrm[e]*zin[bt,src[e],c]  + beta*zprev[bt,n,c]
__global__ __launch_bounds__(256)
void prop_kernel(const float* __restrict__ zin, const float* __restrict__ zprev,
                 const int* __restrict__ rowptr, const int* __restrict__ eid,
                 const int* __restrict__ src, const float* __restrict__ nrm,
                 float* __restrict__ out, float alpha, float beta) {
  int n = blockIdx.x * 4 + (threadIdx.x >> 6);
  int c = threadIdx.x & 63;
  if (n >= Nn) return;
  size_t slice = (size_t)blockIdx.y * Nn * Hc;
  const float* z = zin + slice;
  float acc = 0.f;
  int e1 = rowptr[n + 1];
  for (int i = rowptr[n]; i < e1; ++i) {
    int e = eid[i];
    acc = fmaf(nrm[e], z[(size_t)src[e] * Hc + c], acc);
  }
  float res = alpha * acc;
  size_t o = slice + (size_t)n * Hc + c;
  if (zprev) res = fmaf(beta, zprev[o], res);
  out[o] = res;
}

// ------------------------------------------------- gated temporal conv (WMMA)
// out[b,t,n,co] = relu( c1*sigmoid(c2) + c3 ),  cj = bias_j[co] +
//   sum_{dt,ci} wj[co,ci,dt] * in[b,t+dt,n,ci]
template <int CIN>
__global__ __launch_bounds__(256)
void tconv_kernel(const float* __restrict__ in,
                  const float* __restrict__ w1, const float* __restrict__ b1,
                  const float* __restrict__ w2, const float* __restrict__ b2,
                  const float* __restrict__ w3, const float* __restrict__ b3,
                  float* __restrict__ out, int Tout) {
  __shared__ float ldsB[3 * CIN * Hc];   // (dt*CIN+ci)*64 + co ; <= 48 KB
  const float* wts[3] = {w1, w2, w3};
  const float* bis[3] = {b1, b2, b3};

  const int wave = threadIdx.x >> 5;
  const int lane = threadIdx.x & 31;
  const int half = lane >> 4;
  const int l16  = lane & 15;
  const long row0 = ((long)blockIdx.x * 8 + wave) * 16;
  const long rl   = row0 + l16;                 // A-row owned by this lane
  const int  bidx = (int)(rl / ((long)Tout * Nn));

  v8f acc[3][4];
#pragma unroll
  for (int j = 0; j < 3; ++j)
#pragma unroll
    for (int nt = 0; nt < 4; ++nt) acc[j][nt] = v8f{};

#pragma unroll
  for (int j = 0; j < 3; ++j) {
    __syncthreads();
    // stage W_j transposed: ldsB[(dt*CIN+ci)*64+co] = w[(co*CIN+ci)*3+dt]
    for (int idx = threadIdx.x; idx < 3 * CIN * Hc; idx += 256) {
      int co = idx & 63, kk = idx >> 6;
      int dt = kk / CIN, ci = kk - dt * CIN;
      ldsB[idx] = wts[j][(co * CIN + ci) * 3 + dt];
    }
    __syncthreads();
#pragma unroll
    for (int dt = 0; dt < 3; ++dt) {
      const float* Arow = in + ((size_t)rl + (size_t)(2 * bidx + dt) * Nn) * CIN + 2 * half;
      for (int k = 0; k < CIN; k += 4) {
        v2f a = *(const v2f*)(Arow + k);
        const float* Bp = &ldsB[(dt * CIN + k + 2 * half) * Hc];
#pragma unroll
        for (int nt = 0; nt < 4; ++nt) {
          v2f bf = {Bp[nt * 16 + l16], Bp[Hc + nt * 16 + l16]};
          acc[j][nt] = wmma4(a, bf, acc[j][nt]);
        }
      }
    }
  }

#pragma unroll
  for (int nt = 0; nt < 4; ++nt) {
    int col = nt * 16 + l16;
    float bb1 = bis[0][col], bb2 = bis[1][col], bb3 = bis[2][col];
#pragma unroll
    for (int v = 0; v < 8; ++v) {
      float c1 = acc[0][nt][v] + bb1;
      float c2 = acc[1][nt][v] + bb2;
      float c3 = acc[2][nt][v] + bb3;
      float g  = fmaf(c1, 1.f / (1.f + __expf(-c2)), c3);
      long rowm = row0 + v + 8 * half;
      out[rowm * Hc + col] = fmaxf(g, 0.f);
    }
  }
}

// --------------------------------------------------- Chebyshev GEMM (WMMA)
// out[r,:] = relu( bias + sum_kt Tx_kt[r,:] @ W[kt] )  — row-wise, in-place over tx0
__global__ __launch_bounds__(256)
void cheb_gemm_kernel(const float* __restrict__ tx0, const float* __restrict__ tx1,
                      const float* __restrict__ tx2,
                      const float* __restrict__ W /*(3,64,64) ci,co*/,
                      const float* __restrict__ bias, float* __restrict__ out) {
  __shared__ float ldsB[3 * Hc * Hc];   // direct copy, 48 KB
  for (int idx = threadIdx.x; idx < 3 * Hc * Hc; idx += 256) ldsB[idx] = W[idx];
  __syncthreads();

  const int wave = threadIdx.x >> 5;
  const int lane = threadIdx.x & 31;
  const int half = lane >> 4;
  const int l16  = lane & 15;
  const long row0 = ((long)blockIdx.x * 8 + wave) * 16;

  v8f acc[4];
#pragma unroll
  for (int nt = 0; nt < 4; ++nt) acc[nt] = v8f{};

  const float* srcs[3] = {tx0, tx1, tx2};
#pragma unroll
  for (int kt = 0; kt < 3; ++kt) {
    const float* Arow = srcs[kt] + (size_t)(row0 + l16) * Hc + 2 * half;
    for (int k = 0; k < Hc; k += 4) {
      v2f a = *(const v2f*)(Arow + k);
      const float* Bp = &ldsB[(kt * Hc + k + 2 * half) * Hc];
#pragma unroll
      for (int nt = 0; nt < 4; ++nt) {
        v2f bf = {Bp[nt * 16 + l16], Bp[Hc + nt * 16 + l16]};
        acc[nt] = wmma4(a, bf, acc[nt]);
      }
    }
  }

#pragma unroll
  for (int nt = 0; nt < 4; ++nt) {
    int col = nt * 16 + l16;
    float bb = bias[col];
#pragma unroll
    for (int v = 0; v < 8; ++v) {
      long rowm = row0 + v + 8 * half;
      out[rowm * Hc + col] = fmaxf(acc[nt][v] + bb, 0.f);
    }
  }
}

// ------------------------------------------------ per-node BatchNorm stats
__global__ __launch_bounds__(256)
void bn_stats_kernel(const float* __restrict__ h2, const float* __restrict__ gamma,
                     const float* __restrict__ beta, float* __restrict__ aArr,
                     float* __restrict__ bArr) {
  const int n = blockIdx.x;
  const int BT = Bn * T2;   // 160
  float s = 0.f, ss = 0.f;
  for (int idx = threadIdx.x; idx < BT * Hc; idx += 256) {
    int bt = idx >> 6, c = idx & 63;
    float v = h2[((size_t)bt * Nn + n) * Hc + c];
    s += v; ss = fmaf(v, v, ss);
  }
  __shared__ float sh[256], sh2[256];
  sh[threadIdx.x] = s; sh2[threadIdx.x] = ss;
  __syncthreads();
  for (int off = 128; off > 0; off >>= 1) {
    if (threadIdx.x < off) {
      sh[threadIdx.x]  += sh[threadIdx.x + off];
      sh2[threadIdx.x] += sh2[threadIdx.x + off];
    }
    __syncthreads();
  }
  if (threadIdx.x == 0) {
    float inv_cnt = 1.f / (float)(BT * Hc);
    float mu  = sh[0] * inv_cnt;
    float var = sh2[0] * inv_cnt - mu * mu;
    float a   = gamma[n] * rsqrtf(var + 1e-5f);
    aArr[n] = a;
    bArr[n] = beta[n] - mu * a;   // folded BN shift
  }
}

// -------------------------------------------- BN + linear head, fused
__global__ void final_kernel(const float* __restrict__ h2, const float* __restrict__ aArr,
                             const float* __restrict__ bArr, const float* __restrict__ lw,
                             const float* __restrict__ lb, float* __restrict__ out) {
  long r = (long)blockIdx.x * blockDim.x + threadIdx.x;
  if (r >= M2) return;
  int n = (int)(r % Nn);
  const float* row = h2 + r * Hc;
  float dot = 0.f, sw = 0.f;
#pragma unroll 8
  for (int c = 0; c < Hc; ++c) { float w = lw[c]; dot = fmaf(row[c], w, dot); sw += w; }
  out[r] = fmaf(aArr[n], dot, fmaf(bArr[n], sw, lb[0]));
}

// ============================================================== launcher
extern "C" void kernel_launch(void* const* d_in, const int* in_sizes, int n_in,
                              void* d_out, int out_size, void* d_ws, size_t ws_size,
                              hipStream_t stream) {
  const float* x   = (const float*)d_in[0];
  const int*   ei  = (const int*)d_in[1];
  const float* ew  = (const float*)d_in[2];
  // params1 (indices 3..18) is dead compute in the reference -> skipped.
  const float* t1w1 = (const float*)d_in[19];
  const float* t1b1 = (const float*)d_in[20];
  const float* t1w2 = (const float*)d_in[21];
  const float* t1b2 = (const float*)d_in[22];
  const float* t1w3 = (const float*)d_in[23];
  const float* t1b3 = (const float*)d_in[24];
  const float* t2w1 = (const float*)d_in[25];
  const float* t2b1 = (const float*)d_in[26];
  const float* t2w2 = (const float*)d_in[27];
  const float* t2b2 = (const float*)d_in[28];
  const float* t2w3 = (const float*)d_in[29];
  const float* t2b3 = (const float*)d_in[30];
  const float* chebW = (const float*)d_in[31];
  const float* chebb = (const float*)d_in[32];
  const float* gamma = (const float*)d_in[33];
  const float* beta  = (const float*)d_in[34];
  const float* lw  = (const float*)d_in[35];
  const float* lb  = (const float*)d_in[36];
  const int* srcI = ei;
  const int* dstI = ei + Ec;

  float* ws0 = (float*)d_ws;          // h1 / Tx0 / cheb-relu (in-place)
  float* ws1 = ws0 + S1;              // Tx1, then h2
  float* ws2 = ws0 + 2 * S1;          // Tx2
  float* nrm    = ws0 + 3 * S1;       // E
  float* deg    = nrm + Ec;           // N
  int*   rowptr = (int*)(deg + Nn);   // N+1
  int*   cnt    = rowptr + (Nn + 1);  // N (count, then cursor)
  int*   eid    = cnt + Nn;           // E
  float* aArr   = (float*)(eid + Ec); // N
  float* bArr   = aArr + Nn;          // N

  const int EB = (Ec + 255) / 256;
  const int NB = (Nn + 255) / 256;

  // 1) cheb norm
  zero_f_kernel<<<NB, 256, 0, stream>>>(deg, Nn);
  deg_kernel<<<EB, 256, 0, stream>>>(srcI, ew, deg);
  norm_kernel<<<EB, 256, 0, stream>>>(srcI, dstI, ew, deg, nrm);

  // 2) CSR over dst
  zero_i_kernel<<<NB, 256, 0, stream>>>(cnt, Nn);
  count_kernel<<<EB, 256, 0, stream>>>(dstI, cnt);
  scan_kernel<<<1, 32, 0, stream>>>(cnt, rowptr);
  zero_i_kernel<<<NB, 256, 0, stream>>>(cnt, Nn);
  fill_kernel<<<EB, 256, 0, stream>>>(dstI, rowptr, cnt, eid);

  // 3) temporal conv 1: x (B,24,N,8) -> ws0 (B,22,N,64)
  tconv_kernel<Fc><<<(int)(M1 / 128), 256, 0, stream>>>(
      x, t1w1, t1b1, t1w2, t1b2, t1w3, t1b3, ws0, T1);

  // 4) Chebyshev recurrence: Tx1 = L*Tx0 ; Tx2 = 2*L*Tx1 - Tx0
  dim3 pgrid((Nn + 3) / 4, Bn * T1);
  prop_kernel<<<pgrid, 256, 0, stream>>>(ws0, nullptr, rowptr, eid, srcI, nrm, ws1, 1.f, 0.f);
  prop_kernel<<<pgrid, 256, 0, stream>>>(ws1, ws0,     rowptr, eid, srcI, nrm, ws2, 2.f, -1.f);

  // 5) cheb GEMM + bias + relu, in-place into ws0
  cheb_gemm_kernel<<<(int)(M1 / 128), 256, 0, stream>>>(ws0, ws1, ws2, chebW, chebb, ws0);

  // 6) temporal conv 2: ws0 (B,22,N,64) -> ws1 (B,20,N,64)
  tconv_kernel<Hc><<<(int)(M2 / 128), 256, 0, stream>>>(
      ws0, t2w1, t2b1, t2w2, t2b2, t2w3, t2b3, ws1, T2);

  // 7) per-node BN stats (folded)
  bn_stats_kernel<<<Nn, 256, 0, stream>>>(ws1, gamma, beta, aArr, bArr);

  // 8) BN + linear head
  final_kernel<<<(int)((M2 + 255) / 256), 256, 0, stream>>>(ws1, aArr, bArr, lw, lb, (float*)d_out);
}